// HungarianLoss_53901839564820
// MI455X (gfx1250) — compile-verified
//
#include <hip/hip_runtime.h>
#include <math.h>

#define B_ 64
#define M_ 128        // predictions per batch
#define N_ 64         // target slots per batch
#define CSTRIDE 132   // padded row stride (dwords) for the transposed cost matrix
#define BCE_CLAMP_ 100.0f
#define INF_ 3.0e38f

typedef __attribute__((ext_vector_type(2))) float v2f;
typedef __attribute__((ext_vector_type(8))) float v8f;

// One workgroup per batch: WMMA cost matrix -> wave-parallel Jonker-Volgenant
// -> loss partial sums accumulated into acc[0..3] = {coord, conf, unmatched, matches}.
__global__ __launch_bounds__(128) void HungarianLoss_batch_kernel(
    const float* __restrict__ pred, const float* __restrict__ target,
    float* __restrict__ acc) {
  __shared__ float s_px[M_], s_py[M_], s_pc[M_], s_pn2[M_];
  __shared__ float s_tx[N_], s_ty[N_], s_tn2[N_];
  __shared__ float s_rtx[N_], s_rty[N_];
  __shared__ int   s_rv[N_];
  __shared__ int   s_K;
  __shared__ float s_cost[N_ * CSTRIDE];   // cost[tgt][pred] (transposed)
  __shared__ float s_u[N_ + 1];
  __shared__ int   s_p[M_ + 1];
  __shared__ int   s_way[M_ + 1];

  const int b    = blockIdx.x;
  const int tid  = threadIdx.x;
  const int lane = tid & 31;
  const int wave = tid >> 5;

  // ---- stage loads into LDS ----
  {
    const float* pp = pred + ((size_t)b * M_ + tid) * 3;
    float x = pp[0], y = pp[1], c = pp[2];
    s_px[tid] = x; s_py[tid] = y; s_pc[tid] = c;
    s_pn2[tid] = x * x + y * y;
  }
  if (tid < N_) {
    const float* tp = target + ((size_t)b * N_ + tid) * 3;
    float x = tp[0], y = tp[1], c = tp[2];
    s_rtx[tid] = x; s_rty[tid] = y; s_rv[tid] = (c > 0.5f) ? 1 : 0;
  }
  __syncthreads();
  if (tid == 0) {      // compact valid targets
    int k = 0;
    for (int t = 0; t < N_; ++t)
      if (s_rv[t]) { s_tx[k] = s_rtx[t]; s_ty[k] = s_rty[t]; ++k; }
    s_K = k;
    for (int t = k; t < N_; ++t) { s_tx[t] = 0.0f; s_ty[t] = 0.0f; }
  }
  __syncthreads();
  if (tid < N_) {
    float x = s_tx[tid], y = s_ty[tid];
    s_tn2[tid] = x * x + y * y;
  }
  __syncthreads();

  // ---- cost matrix via WMMA: dot-product cross term, K=2 padded to 4 ----
  {
    const int half = lane >> 4;   // A/B: lanes 0-15 carry K=0,1; lanes 16-31 K=2,3 (zero)
    const int l16  = lane & 15;
    for (int rt = 2 * wave; rt < 2 * wave + 2; ++rt) {     // pred row tiles
      const int m = rt * 16 + l16;
      v2f a;
      a.x = (half == 0) ? s_px[m] : 0.0f;
      a.y = (half == 0) ? s_py[m] : 0.0f;
      for (int ct = 0; ct < 4; ++ct) {                     // target col tiles
        const int n = ct * 16 + l16;
        v2f bb;
        bb.x = (half == 0) ? s_tx[n] : 0.0f;
        bb.y = (half == 0) ? s_ty[n] : 0.0f;
        v8f c = {};
#if __has_builtin(__builtin_amdgcn_wmma_f32_16x16x4_f32)
        c = __builtin_amdgcn_wmma_f32_16x16x4_f32(
                false, a, false, bb, (short)0, c, false, false);
        #pragma unroll
        for (int r = 0; r < 8; ++r) {
          int mm = rt * 16 + r + 8 * half;    // C/D layout: vgpr r -> M=r (+8 for hi half)
          int nn = ct * 16 + l16;
          float d2 = s_pn2[mm] + s_tn2[nn] - 2.0f * c[r];
          s_cost[nn * CSTRIDE + mm] = sqrtf(fmaxf(d2, 0.0f));
        }
#else
        // fallback: direct distance (no WMMA available)
        #pragma unroll
        for (int r = 0; r < 8; ++r) {
          int mm = rt * 16 + r + 8 * half;
          int nn = ct * 16 + l16;
          float dx = s_px[mm] - s_tx[nn], dy = s_py[mm] - s_ty[nn];
          s_cost[nn * CSTRIDE + mm] = sqrtf(dx * dx + dy * dy);
        }
#endif
      }
    }
  }
  __syncthreads();

  if (tid >= 32) return;   // wave 0 carries on: JV matching + loss. No more block barriers.

  const int K = s_K;

  // init potentials / matching state (lane-partitioned)
  for (int i = lane; i <= N_; i += 32) s_u[i] = 0.0f;
  float v_r[5], minv[5];
  #pragma unroll
  for (int t = 0; t < 5; ++t) v_r[t] = 0.0f;
  #pragma unroll
  for (int t = 0; t < 5; ++t) {
    int j = t * 32 + lane;
    if (j <= M_) { s_p[j] = 0; s_way[j] = 0; }
  }

  // Jonker-Volgenant on transposed matrix: rows = K targets, cols = 128 preds
  for (int i = 1; i <= K; ++i) {
    if (lane == 0) s_p[0] = i;
    int j0 = 0;
    unsigned used_mask = 0;
    #pragma unroll
    for (int t = 0; t < 5; ++t) minv[t] = INF_;

    for (;;) {
      #pragma unroll
      for (int t = 0; t < 5; ++t)
        if (t * 32 + lane == j0) used_mask |= (1u << t);
      const int   i0  = s_p[j0];
      const float ui0 = s_u[i0];

      float best = INF_;
      int bestj = 0;
      #pragma unroll
      for (int t = 0; t < 5; ++t) {
        int j = t * 32 + lane;
        if (j >= 1 && j <= M_ && !(used_mask & (1u << t))) {
          float cur = s_cost[(i0 - 1) * CSTRIDE + (j - 1)] - ui0 - v_r[t];
          if (cur < minv[t]) { minv[t] = cur; s_way[j] = j0; }
          if (minv[t] < best) { best = minv[t]; bestj = j; }
        }
      }
      // wave32 argmin reduction (match argmin-first tie-break: lower j wins)
      #pragma unroll
      for (int off = 16; off >= 1; off >>= 1) {
        float ob = __shfl_xor(best, off, 32);
        int   oj = __shfl_xor(bestj, off, 32);
        if (ob < best || (ob == best && oj != 0 && (bestj == 0 || oj < bestj))) {
          best = ob; bestj = oj;
        }
      }
      const float delta = best;
      const int   j1    = bestj;

      #pragma unroll
      for (int t = 0; t < 5; ++t) {
        int j = t * 32 + lane;
        if (j <= M_) {
          if (used_mask & (1u << t)) {
            int pj = s_p[j];              // distinct rows per used col -> no collision
            s_u[pj] += delta;
            v_r[t]  -= delta;
          } else {
            minv[t] -= delta;
          }
        }
      }
      j0 = j1;
      if (s_p[j0] == 0) break;
    }
    if (lane == 0) {                      // augmenting-path flip (short serial walk)
      int j = j0;
      while (j) {
        int jw = s_way[j];
        s_p[j] = s_p[jw];
        j = jw;
      }
    }
  }

  // ---- loss partials ----
  float coord = 0.0f, conf = 0.0f, unmatched = 0.0f;
  const float uw = (K > 0) ? 1.0f : (1.0f / (float)M_);
  #pragma unroll
  for (int t = 0; t < 4; ++t) {
    int j    = t * 32 + lane + 1;         // 1..128
    int pi   = (K > 0) ? s_p[j] : 0;
    int pidx = j - 1;
    float pc = s_pc[pidx];
    if (pi != 0) {
      int ti = pi - 1;                    // compacted target index (coords already gathered)
      float dx = s_px[pidx] - s_tx[ti];
      float dy = s_py[pidx] - s_ty[ti];
      coord += 0.5f * (dx * dx + dy * dy);
      conf  += fminf(-logf(pc), BCE_CLAMP_);
    } else {
      unmatched += uw * fminf(-log1pf(-pc), BCE_CLAMP_);
    }
  }
  #pragma unroll
  for (int off = 16; off >= 1; off >>= 1) {
    coord     += __shfl_xor(coord, off, 32);
    conf      += __shfl_xor(conf, off, 32);
    unmatched += __shfl_xor(unmatched, off, 32);
  }
  if (lane == 0) {
    atomicAdd(&acc[0], coord);
    atomicAdd(&acc[1], conf);
    atomicAdd(&acc[2], unmatched);
    atomicAdd(&acc[3], (float)K);
  }
}

__global__ void HungarianLoss_zero_kernel(float* acc) {
  if (threadIdx.x < 4) acc[threadIdx.x] = 0.0f;
}

__global__ void HungarianLoss_finalize_kernel(const float* __restrict__ acc,
                                              float* __restrict__ out) {
  float total = acc[3];
  float s = acc[0] + acc[1] + acc[2];
  out[0] = (total > 0.0f) ? (s / total) : s;
}

extern "C" void kernel_launch(void* const* d_in, const int* in_sizes, int n_in,
                              void* d_out, int out_size, void* d_ws, size_t ws_size,
                              hipStream_t stream) {
  (void)in_sizes; (void)n_in; (void)out_size; (void)ws_size;
  const float* pred   = (const float*)d_in[0];   // (64,128,3) f32
  const float* target = (const float*)d_in[1];   // (64,64,3)  f32
  float* out = (float*)d_out;                    // scalar f32
  float* acc = (float*)d_ws;                     // 4 floats of scratch

  HungarianLoss_zero_kernel<<<1, 32, 0, stream>>>(acc);
  HungarianLoss_batch_kernel<<<B_, 128, 0, stream>>>(pred, target, acc);
  HungarianLoss_finalize_kernel<<<1, 1, 0, stream>>>(acc, out);
}